// TreeAttention_31086973288483
// MI455X (gfx1250) — compile-verified
//
#include <hip/hip_runtime.h>
#include <math.h>
#include <cstdint>

typedef __attribute__((ext_vector_type(2))) float v2f;
typedef __attribute__((ext_vector_type(8))) float v8f;
typedef int v4i __attribute__((vector_size(16)));

#if defined(__has_builtin)
#  if __has_builtin(__builtin_amdgcn_global_load_async_to_lds_b128)
#    define HAVE_ASYNC_LDS 1
#  endif
#endif
#ifndef HAVE_ASYNC_LDS
#  define HAVE_ASYNC_LDS 0
#endif

typedef __attribute__((address_space(1))) v4i* g4_t;  // global int4*
typedef __attribute__((address_space(3))) v4i* l4_t;  // LDS int4* (32-bit)

namespace {
constexpr int kB = 2, kS = 2048, kE = 512, kH = 8, kD = 64;
constexpr int kM = kB * kS;       // 4096 token rows
constexpr int kWaveN = 64;        // wave computes 16 x 64 tile (4 WMMA tiles on N)
constexpr int kBlockM = 128;      // 8 waves stacked along M
constexpr int KC = 32;            // K chunk staged in LDS
constexpr int NCHUNK = kE / KC;   // 16
}

template <int N>
__device__ __forceinline__ void wait_async_le() {
#if HAVE_ASYNC_LDS
#  if __has_builtin(__builtin_amdgcn_s_wait_asynccnt)
  __builtin_amdgcn_s_wait_asynccnt(N);
#  else
  asm volatile("s_wait_asynccnt %0" ::"n"(N) : "memory");
#  endif
#endif
}

// ---------------------------------------------------------------------------
// Stage one 32(K) x 64(N) fp32 chunk of W into LDS (8 KB), block-cooperative.
// 256 threads x 2 float4 each.  LDS layout: row-major 32x64 with XOR-16
// column swizzle keyed on (row>>1)&1 so the two lane halves of a wave (which
// read rows K and K+2) hit disjoint bank groups during compute.
// ---------------------------------------------------------------------------
__device__ __forceinline__ void stage_chunk(const float* __restrict__ W,
                                            float* __restrict__ buf,
                                            int k0, int Nbase, int t)
{
#pragma unroll
  for (int i = 0; i < 2; ++i) {
    const int q    = t + (i << 8);            // 0..511 linear float4 id
    const int r    = q >> 4;                  // chunk row 0..31
    const int col  = (q & 15) << 2;           // 0,4,...,60
    const int pcol = col ^ (((r >> 1) & 1) << 4);
    const float* g = W + (size_t)(k0 + r) * kE + Nbase + col;
    float* l = buf + (r << 6) + pcol;
#if HAVE_ASYNC_LDS
    __builtin_amdgcn_global_load_async_to_lds_b128(
        (g4_t)(unsigned long long)(uintptr_t)g,
        (l4_t)(unsigned int)(uintptr_t)l, 0, 0);
#else
    *(float4*)l = *(const float4*)g;
#endif
  }
}

// ---------------------------------------------------------------------------
// Per-wave compute over one staged K chunk.
// A-frag: lane l -> row m = l%16, K pair 2*(l/16); vgpr v -> K+v (global b64).
// B-frag: from LDS with the XOR-16 swizzle (conflict-free between halves).
// ---------------------------------------------------------------------------
__device__ __forceinline__ void compute_chunk(
    const float* __restrict__ arow, const float* __restrict__ buf,
    int kglob, int lmod, int lhi,
    v8f& acc0, v8f& acc1, v8f& acc2, v8f& acc3)
{
#pragma unroll
  for (int kk = 0; kk < KC; kk += 4) {
    const int ka = kglob + kk + 2 * lhi;
    const float2 af = *(const float2*)(arow + ka);   // contiguous 8B load
    v2f a; a.x = af.x; a.y = af.y;

    const int r0 = kk + 2 * lhi;                     // LDS rows r0, r0+1
    const int js = (r0 >> 1) & 1;                    // swizzle group select
    const float* row0 = buf + (r0 << 6);
    const float* row1 = row0 + 64;

    v2f b0; b0.x = row0[lmod + 16 * (0 ^ js)]; b0.y = row1[lmod + 16 * (0 ^ js)];
    v2f b1; b1.x = row0[lmod + 16 * (1 ^ js)]; b1.y = row1[lmod + 16 * (1 ^ js)];
    v2f b2; b2.x = row0[lmod + 16 * (2 ^ js)]; b2.y = row1[lmod + 16 * (2 ^ js)];
    v2f b3; b3.x = row0[lmod + 16 * (3 ^ js)]; b3.y = row1[lmod + 16 * (3 ^ js)];

    acc0 = __builtin_amdgcn_wmma_f32_16x16x4_f32(false, a, false, b0, (short)0, acc0, false, false);
    acc1 = __builtin_amdgcn_wmma_f32_16x16x4_f32(false, a, false, b1, (short)0, acc1, false, false);
    acc2 = __builtin_amdgcn_wmma_f32_16x16x4_f32(false, a, false, b2, (short)0, acc2, false, false);
    acc3 = __builtin_amdgcn_wmma_f32_16x16x4_f32(false, a, false, b3, (short)0, acc3, false, false);
  }
}

// ---------------------------------------------------------------------------
// Block GEMM body: O[Mb:Mb+128, Nb:Nb+64] = A * W + bias with LDS-staged W,
// double-buffered K chunks (async global->LDS when available).
// ---------------------------------------------------------------------------
__device__ __forceinline__ void gemm_block(
    const float* __restrict__ A, const float* __restrict__ W,
    const float* __restrict__ bias, float* __restrict__ O,
    int MbaseBlk, int Nbase, float* __restrict__ wbuf /* [2][KC*64] */)
{
  const int t    = threadIdx.x;
  const int wave = t >> 5;
  const int lane = t & 31;
  const int lmod = lane & 15;
  const int lhi  = lane >> 4;

  const int Mbase = MbaseBlk + wave * 16;
  const float* __restrict__ arow = A + (size_t)(Mbase + lmod) * kE;
  const int n0 = Nbase + lmod;

  v8f acc0 = {}, acc1 = {}, acc2 = {}, acc3 = {};

  stage_chunk(W, wbuf, 0, Nbase, t);

  for (int c = 0; c < NCHUNK; ++c) {
    if (c + 1 < NCHUNK) {
      stage_chunk(W, wbuf + ((c + 1) & 1) * (KC * 64), (c + 1) * KC, Nbase, t);
      __builtin_prefetch(arow + (c + 1) * KC, 0, 0);  // global_prefetch_b8 for A
      wait_async_le<2>();   // current chunk's 2 async ops landed; next 2 in flight
    } else {
      wait_async_le<0>();
    }
    __syncthreads();
    compute_chunk(arow, wbuf + (c & 1) * (KC * 64), c * KC, lmod, lhi,
                  acc0, acc1, acc2, acc3);
    __syncthreads();        // protect buffer reuse before next stage overwrites
  }

  // Epilogue: add bias, scatter-store D fragments.
  const float bi0 = bias[n0];
  const float bi1 = bias[n0 + 16];
  const float bi2 = bias[n0 + 32];
  const float bi3 = bias[n0 + 48];
  const int mrow = Mbase + 8 * lhi;
#pragma unroll
  for (int v = 0; v < 8; ++v) {
    float* __restrict__ orow = O + (size_t)(mrow + v) * kE + n0;
    orow[0]  = acc0[v] + bi0;
    orow[16] = acc1[v] + bi1;
    orow[32] = acc2[v] + bi2;
    orow[48] = acc3[v] + bi3;
  }
}

// ---------------------------------------------------------------------------
// Kernel 1: fused tri-projection. blockIdx.z selects {Wp,bp}->P, {Wc,bc}->C,
// {Ws,bs}->S.  grid = (kE/64, kM/128, 3), block = 256 (8 waves).
// ---------------------------------------------------------------------------
__global__ __launch_bounds__(256) void tri_proj_kernel(
    const float* __restrict__ X,
    const float* __restrict__ Wp, const float* __restrict__ bp,
    const float* __restrict__ Wc, const float* __restrict__ bc,
    const float* __restrict__ Ws, const float* __restrict__ bs,
    float* __restrict__ P, float* __restrict__ C, float* __restrict__ S)
{
  __shared__ float wbuf[2][KC * 64];
  const float* W; const float* bia; float* O;
  if (blockIdx.z == 0)      { W = Wp; bia = bp; O = P; }
  else if (blockIdx.z == 1) { W = Wc; bia = bc; O = C; }
  else                      { W = Ws; bia = bs; O = S; }
  gemm_block(X, W, bia, O, blockIdx.y * kBlockM, blockIdx.x * kWaveN,
             &wbuf[0][0]);
}

// ---------------------------------------------------------------------------
// Kernel 3: output projection AO * Wo + bo -> OUT.
// ---------------------------------------------------------------------------
__global__ __launch_bounds__(256) void out_proj_kernel(
    const float* __restrict__ AO, const float* __restrict__ Wo,
    const float* __restrict__ bo, float* __restrict__ OUT)
{
  __shared__ float wbuf[2][KC * 64];
  gemm_block(AO, Wo, bo, OUT, blockIdx.y * kBlockM, blockIdx.x * kWaveN,
             &wbuf[0][0]);
}

// ---------------------------------------------------------------------------
// Kernel 2: sparse tree attention with online softmax.
// One thread per (b, n, h); 64-thread blocks (2 waves) share the parent row
// of batch b in LDS (blocks never straddle a batch: 16384 % 64 == 0).
// Each row n attends only to {parent[n]} U {m : parent[m] == n}.
// ---------------------------------------------------------------------------
__global__ __launch_bounds__(64) void tree_attn_kernel(
    const float* __restrict__ P, const float* __restrict__ C,
    const float* __restrict__ S, const int* __restrict__ parent,
    float* __restrict__ AO)
{
  __shared__ int sp[kS];

  const int t   = threadIdx.x;
  const int gid = blockIdx.x * 64 + t;       // 0 .. B*S*H-1
  const int b   = gid / (kS * kH);

  const int* __restrict__ prow_g = parent + b * kS;
  for (int i = t; i < kS; i += 64) sp[i] = prow_g[i];
  __syncthreads();

  const int nh = gid % (kS * kH);
  const int n  = nh / kH;
  const int h  = nh % kH;

  const size_t rowbase = (size_t)(b * kS) * kE + h * kD;
  const float* __restrict__ pv_g = P + rowbase + (size_t)n * kE;

  float pv[kD];
#pragma unroll
  for (int d = 0; d < kD; d += 4) {
    const float4 q = *(const float4*)(pv_g + d);
    pv[d] = q.x; pv[d + 1] = q.y; pv[d + 2] = q.z; pv[d + 3] = q.w;
  }

  float acc[kD];
#pragma unroll
  for (int d = 0; d < kD; ++d) acc[d] = 0.0f;
  float mmax = -INFINITY;
  float ssum = 0.0f;

  const int mp = sp[n];
  for (int m = 0; m < kS; ++m) {
    const bool match = (m == mp) || (sp[m] == n);
    if (!match) continue;

    const float* __restrict__ crow = C + rowbase + (size_t)m * kE;
    float sc = 0.0f;
#pragma unroll
    for (int d = 0; d < kD; d += 4) {
      const float4 cq = *(const float4*)(crow + d);
      sc += pv[d] * cq.x + pv[d + 1] * cq.y + pv[d + 2] * cq.z + pv[d + 3] * cq.w;
    }
    sc *= 0.125f;   // 1/sqrt(64)

    const float nm   = fmaxf(mmax, sc);
    const float corr = __expf(mmax - nm);   // exp(-inf - finite) = 0 on first hit
    const float e    = __expf(sc - nm);
    ssum = ssum * corr + e;

    const float* __restrict__ srow = S + rowbase + (size_t)m * kE;
#pragma unroll
    for (int d = 0; d < kD; d += 4) {
      const float4 sq = *(const float4*)(srow + d);
      acc[d]     = acc[d]     * corr + e * sq.x;
      acc[d + 1] = acc[d + 1] * corr + e * sq.y;
      acc[d + 2] = acc[d + 2] * corr + e * sq.z;
      acc[d + 3] = acc[d + 3] * corr + e * sq.w;
    }
    mmax = nm;
  }

  // Every row has at least its parent edge -> ssum > 0.
  const float inv = 1.0f / ssum;
  float* __restrict__ orow = AO + rowbase + (size_t)n * kE;
#pragma unroll
  for (int d = 0; d < kD; d += 4) {
    float4 o;
    o.x = acc[d] * inv; o.y = acc[d + 1] * inv;
    o.z = acc[d + 2] * inv; o.w = acc[d + 3] * inv;
    *(float4*)(orow + d) = o;
  }
}

// ---------------------------------------------------------------------------
// Launch: tri-projection -> sparse attention -> output projection.
// Workspace: P | C | S | AO, each kM*kE fp32 (8 MB), total 32 MB.
// ---------------------------------------------------------------------------
extern "C" void kernel_launch(void* const* d_in, const int* in_sizes, int n_in,
                              void* d_out, int out_size, void* d_ws, size_t ws_size,
                              hipStream_t stream) {
  const float* x      = (const float*)d_in[0];
  const int*   parent = (const int*)d_in[1];
  const float* Wp     = (const float*)d_in[2];
  const float* bp     = (const float*)d_in[3];
  const float* Wc     = (const float*)d_in[4];
  const float* bc     = (const float*)d_in[5];
  const float* Ws     = (const float*)d_in[6];
  const float* bs     = (const float*)d_in[7];
  const float* Wo     = (const float*)d_in[8];
  const float* bo     = (const float*)d_in[9];
  float* out = (float*)d_out;

  const size_t mat = (size_t)kM * kE;
  float* P  = (float*)d_ws;
  float* C  = P + mat;
  float* S  = C + mat;
  float* AO = S + mat;

  {
    dim3 grid(kE / kWaveN, kM / kBlockM, 3);   // (8, 32, 3)
    tri_proj_kernel<<<grid, dim3(256), 0, stream>>>(x, Wp, bp, Wc, bc, Ws, bs,
                                                    P, C, S);
  }
  {
    dim3 grid((kB * kS * kH) / 64);            // 512 blocks of 64 threads
    tree_attn_kernel<<<grid, dim3(64), 0, stream>>>(P, C, S, parent, AO);
  }
  {
    dim3 grid(kE / kWaveN, kM / kBlockM, 1);   // (8, 32)
    out_proj_kernel<<<grid, dim3(256), 0, stream>>>(AO, Wo, bo, out);
  }
}